// HarmonicResNetBlock_29205777613309
// MI455X (gfx1250) — compile-verified
//
#include <hip/hip_runtime.h>

typedef __attribute__((ext_vector_type(2))) float v2f;
typedef __attribute__((ext_vector_type(8))) float v8f;

static constexpr int   N_NODES = 10000;
static constexpr int   N_EDGES = 160000;
static constexpr float EPSF    = 1e-8f;

// ---- workspace layout (in floats) ----
static constexpr size_t SZ_AGG1 = (size_t)N_NODES * 256;           // (N, o=2, t=2, K=64)
static constexpr size_t SZ_AGG2 = (size_t)N_NODES * 512;           // (N, o=2, t=2, K=128)
static constexpr size_t SZ_H    = (size_t)N_NODES * 2 * 32 * 2;    // (N, m=2, 32, 2)
static constexpr size_t SZ_W1   = 2 * 64 * 32;                     // (o, K/2, p, 2)
static constexpr size_t SZ_W2   = 2 * 128 * 32;
static constexpr size_t OFF_AGG1 = 0;
static constexpr size_t OFF_AGG2 = OFF_AGG1 + SZ_AGG1;
static constexpr size_t OFF_H    = OFF_AGG2 + SZ_AGG2;
static constexpr size_t OFF_W1R  = OFF_H + SZ_H;
static constexpr size_t OFF_W1I  = OFF_W1R + SZ_W1;
static constexpr size_t OFF_W1N  = OFF_W1I + SZ_W1;
static constexpr size_t OFF_W2R  = OFF_W1N + SZ_W1;
static constexpr size_t OFF_W2I  = OFF_W2R + SZ_W2;
static constexpr size_t OFF_W2N  = OFF_W2I + SZ_W2;

__device__ __forceinline__ void atomic_add_f32(float* p, float v) {
#if defined(__HIP_DEVICE_COMPILE__)
    unsafeAtomicAdd(p, v);   // native global_atomic_add_f32
#else
    atomicAdd(p, v);
#endif
}

// --------------------------------------------------------------------------
// Bake complex weights in B-fragment-packed order:
//   W*(o, k, p) at index (((o*(K/2) + (k>>1))*32 + p)*2 + (k&1))
//   Wr = rad*cos(ph), Wi = rad*sin(ph), Wn = -Wi
//   k = m*(2C) + r*C + i ; K = 4C ; s = m*2 + o
// --------------------------------------------------------------------------
template <int C>
__global__ void setup_w(const float* __restrict__ radial,
                        const float* __restrict__ phase,
                        float* __restrict__ Wr, float* __restrict__ Wi,
                        float* __restrict__ Wn) {
    constexpr int K = 4 * C;
    int tid = blockIdx.x * blockDim.x + threadIdx.x;
    if (tid >= 2 * K * 32) return;
    int p  = tid % 32;
    int km = (tid / 32) % K;
    int o  = tid / (32 * K);
    int m   = km / (2 * C);
    int rem = km % (2 * C);
    int r   = rem / C;
    int i   = rem % C;
    int s   = m * 2 + o;
    float rad = radial[((size_t)(s * 2 + r) * 32 + p) * C + i];
    float ph  = phase[((size_t)s * 32 + p) * C + i];
    size_t idx = (((size_t)o * (K / 2) + (km >> 1)) * 32 + p) * 2 + (km & 1);
    float wr = rad * cosf(ph);
    float wi = rad * sinf(ph);
    Wr[idx] = wr;
    Wi[idx] = wi;
    Wn[idx] = -wi;
}

// --------------------------------------------------------------------------
// Per-edge scatter: msg[m,o,r,i] = cmul(xj[m,i], pmo[m,o,r]) -> atomic agg[src]
// agg layout: ((n*2 + o)*2 + t)*K + (m*2C + r*C + i),  K = 4C
// --------------------------------------------------------------------------
template <int C>
__global__ void edge_scatter(const float* __restrict__ x,
                             const int*   __restrict__ ei,
                             const float* __restrict__ precomp,
                             const float* __restrict__ conn,
                             float* __restrict__ agg) {
    constexpr int K = 4 * C;
    int tid = blockIdx.x * blockDim.x + threadIdx.x;
    if (tid >= N_EDGES * C) return;
    int e = tid / C;
    int i = tid % C;
    int s  = ei[e];            // segment target (sum by src)
    int tg = ei[N_EDGES + e];  // feature gather index

    const float* xb = x + (size_t)tg * 2 * C * 2;
    float x0r = xb[i * 2],       x0i = xb[i * 2 + 1];
    float x1r = xb[(C + i) * 2], x1i = xb[(C + i) * 2 + 1];
    float cr = conn[e * 2], ci = conn[e * 2 + 1];
    float y1r = x1r * cr - x1i * ci;     // c_mul(x[:,1], connection)
    float y1i = x1r * ci + x1i * cr;

    float pre[2][2][2];                  // [ord][r][t]
    const float* pc = precomp + (size_t)e * 8;
#pragma unroll
    for (int q = 0; q < 8; ++q) ((float*)pre)[q] = pc[q];

#pragma unroll
    for (int m = 0; m < 2; ++m) {
        float xr = m ? y1r : x0r;
        float xi = m ? y1i : x0i;
#pragma unroll
        for (int o = 0; o < 2; ++o) {
            // m=0: ord=o ; m=1: o=0 -> conj(precomp[1]), o=1 -> precomp[0]
            int   ord = (m == 0) ? o : 1 - o;
            float sgn = (m == 1 && o == 0) ? -1.0f : 1.0f;
#pragma unroll
            for (int r = 0; r < 2; ++r) {
                float pr = pre[ord][r][0];
                float pi = sgn * pre[ord][r][1];
                float mr = xr * pr - xi * pi;
                float mi = xr * pi + xi * pr;
                size_t base = ((size_t)s * 2 + o) * 2 * K + (size_t)(m * 2 * C + r * C + i);
                atomic_add_f32(&agg[base],     mr);
                atomic_add_f32(&agg[base + K], mi);
            }
        }
    }
}

// --------------------------------------------------------------------------
// fp32 WMMA: D = A(16x4) * B(4x16) + C(16x16)
// --------------------------------------------------------------------------
__device__ __forceinline__ v8f wmma4(v2f a, v2f b, v8f c) {
    return __builtin_amdgcn_wmma_f32_16x16x4_f32(false, a, false, b, (short)0, c,
                                                 false, false);
}

// --------------------------------------------------------------------------
// Per-node complex GEMM (one wave = one 16-node tile, one o-stream):
//   conv[n,o,p] = sum_k A_c[n,k] * W_c[o,k,p]   (K = 64 or 128, P = 32)
// then (FINAL) + residual x*lin_w^T + lin_b, then modReLU with `bias`.
// --------------------------------------------------------------------------
template <int K, bool FINAL>
__global__ void __launch_bounds__(32)
node_conv(const float* __restrict__ agg,
          const float* __restrict__ Wr, const float* __restrict__ Wi,
          const float* __restrict__ Wn,
          const float* __restrict__ bias,
          const float* __restrict__ xres,
          const float* __restrict__ lin_w,
          const float* __restrict__ lin_b,
          float* __restrict__ out) {
    int tile = blockIdx.x;
    int o    = blockIdx.y;
    int lane = threadIdx.x;
    int half = lane >> 4;
    int l16  = lane & 15;

    int rowA = tile * 16 + l16;
    const float* aR = agg + (((size_t)rowA * 2 + o) * 2 + 0) * K + half * 2;
    const float* aI = aR + K;
    // B-fragment packed: frag(kpair, col) at W + (kpair*32 + col)*2 (8B aligned)
    const float* wR = Wr + (size_t)o * (K / 2) * 64;
    const float* wI = Wi + (size_t)o * (K / 2) * 64;
    const float* wN = Wn + (size_t)o * (K / 2) * 64;

    v8f cre0 = {0, 0, 0, 0, 0, 0, 0, 0}, cim0 = {0, 0, 0, 0, 0, 0, 0, 0};
    v8f cre1 = {0, 0, 0, 0, 0, 0, 0, 0}, cim1 = {0, 0, 0, 0, 0, 0, 0, 0};

    for (int k0 = 0; k0 < K; k0 += 4) {
        v2f ar = *(const v2f*)(aR + k0);       // A[l16][k0+2*half .. +1]
        v2f ai = *(const v2f*)(aI + k0);
        size_t kp  = (size_t)(k0 >> 1) + half; // fragment row pair index
        size_t f0  = (kp * 32 + l16) * 2;      // col tile 0 (p = l16)
        size_t f1  = f0 + 32;                  // col tile 1 (p = 16 + l16)
        v2f wr_a  = *(const v2f*)(wR + f0);
        v2f wi_a  = *(const v2f*)(wI + f0);
        v2f nwi_a = *(const v2f*)(wN + f0);
        v2f wr_b  = *(const v2f*)(wR + f1);
        v2f wi_b  = *(const v2f*)(wI + f1);
        v2f nwi_b = *(const v2f*)(wN + f1);

        cre0 = wmma4(ar, wr_a,  cre0);         // Re += Ar*Wr
        cre0 = wmma4(ai, nwi_a, cre0);         // Re -= Ai*Wi
        cim0 = wmma4(ar, wi_a,  cim0);         // Im += Ar*Wi
        cim0 = wmma4(ai, wr_a,  cim0);         // Im += Ai*Wr
        cre1 = wmma4(ar, wr_b,  cre1);
        cre1 = wmma4(ai, nwi_b, cre1);
        cim1 = wmma4(ar, wi_b,  cim1);
        cim1 = wmma4(ai, wr_b,  cim1);
    }

#pragma unroll
    for (int j = 0; j < 8; ++j) {
        int row = j + half * 8;
        int n   = tile * 16 + row;
#pragma unroll
        for (int ct = 0; ct < 2; ++ct) {
            int   col = ct * 16 + l16;
            float re  = ct ? cre1[j] : cre0[j];
            float im  = ct ? cim1[j] : cim0[j];
            if (FINAL) {
                float rr = lin_b[col], ri = lin_b[col];
                const float* xb = xres + ((size_t)n * 2 + o) * 16 * 2;
#pragma unroll
                for (int ii = 0; ii < 16; ++ii) {
                    float w = lin_w[col * 16 + ii];
                    rr += xb[ii * 2]     * w;
                    ri += xb[ii * 2 + 1] * w;
                }
                re += rr;
                im += ri;
            }
            float mag = sqrtf(re * re + im * im + EPSF);
            float rb  = fmaxf(mag + bias[col], 0.0f);
            float scl = rb / mag;
            float2 v  = {re * scl, im * scl};
            *(float2*)(out + (((size_t)n * 2 + o) * 32 + col) * 2) = v;
        }
    }
}

// --------------------------------------------------------------------------
extern "C" void kernel_launch(void* const* d_in, const int* in_sizes, int n_in,
                              void* d_out, int out_size, void* d_ws, size_t ws_size,
                              hipStream_t stream) {
    const float* x       = (const float*)d_in[0];
    const int*   ei      = (const int*)  d_in[1];
    const float* precomp = (const float*)d_in[2];
    const float* conn    = (const float*)d_in[3];
    const float* radial1 = (const float*)d_in[4];
    const float* phase1  = (const float*)d_in[5];
    const float* bias1   = (const float*)d_in[6];
    const float* radial2 = (const float*)d_in[7];
    const float* phase2  = (const float*)d_in[8];
    const float* lin_w   = (const float*)d_in[9];
    const float* lin_b   = (const float*)d_in[10];
    const float* bias2   = (const float*)d_in[11];

    float* ws   = (float*)d_ws;
    float* agg1 = ws + OFF_AGG1;
    float* agg2 = ws + OFF_AGG2;
    float* h    = ws + OFF_H;
    float* W1r  = ws + OFF_W1R;
    float* W1i  = ws + OFF_W1I;
    float* W1n  = ws + OFF_W1N;
    float* W2r  = ws + OFF_W2R;
    float* W2i  = ws + OFF_W2I;
    float* W2n  = ws + OFF_W2N;

    // zero the atomic accumulators (agg1 and agg2 are contiguous)
    hipMemsetAsync(d_ws, 0, (SZ_AGG1 + SZ_AGG2) * sizeof(float), stream);

    setup_w<16><<<(2 * 64 * 32 + 255) / 256, 256, 0, stream>>>(radial1, phase1, W1r, W1i, W1n);
    setup_w<32><<<(2 * 128 * 32 + 255) / 256, 256, 0, stream>>>(radial2, phase2, W2r, W2i, W2n);

    edge_scatter<16><<<(N_EDGES * 16 + 255) / 256, 256, 0, stream>>>(x, ei, precomp, conn, agg1);

    node_conv<64, false><<<dim3(N_NODES / 16, 2), 32, 0, stream>>>(
        agg1, W1r, W1i, W1n, bias1, nullptr, nullptr, nullptr, h);

    edge_scatter<32><<<(N_EDGES * 32 + 255) / 256, 256, 0, stream>>>(h, ei, precomp, conn, agg2);

    node_conv<128, true><<<dim3(N_NODES / 16, 2), 32, 0, stream>>>(
        agg2, W2r, W2i, W2n, bias2, x, lin_w, lin_b, (float*)d_out);
}